// RaWRGCNLayer_81673098101282
// MI455X (gfx1250) — compile-verified
//
#include <hip/hip_runtime.h>

#define NUM_ENTITIES 100000
#define NUM_WALKS    1000000
#define WALK_LEN     3
#define INPUT_DIM    128
#define OUTPUT_DIM   256
#define NUM_REL2     192

typedef __attribute__((ext_vector_type(2))) float v2f;
typedef __attribute__((ext_vector_type(8))) float v8f;

// ---------------------------------------------------------------------------
// Kernel 0: zero a float buffer (accumulators must be re-zeroed every call)
// ---------------------------------------------------------------------------
__global__ void zero_f32(float* __restrict__ p, long long n) {
  long long i = (long long)blockIdx.x * blockDim.x + threadIdx.x;
  if (i < n) p[i] = 0.0f;
}

// ---------------------------------------------------------------------------
// Kernel 1: P[l][r][o] = sum_k rel_emb[r][k] * weight[l*128+k][o]
// Three [192x128] @ [128x256] fp32 GEMMs via V_WMMA_F32_16X16X4_F32.
// One wave (32 lanes) per 16x16 output tile; 3*12*16 = 576 tiles.
// All branching is wave-uniform => EXEC all-ones inside WMMA region.
// ---------------------------------------------------------------------------
__global__ void project_wmma(const float* __restrict__ rel_emb,
                             const float* __restrict__ weight,
                             float* __restrict__ P) {
  const int wave = (blockIdx.x * blockDim.x + threadIdx.x) >> 5;
  const int lane = threadIdx.x & 31;
  if (wave >= WALK_LEN * (NUM_REL2 / 16) * (OUTPUT_DIM / 16)) return;

  const int tiles_n = OUTPUT_DIM / 16;                 // 16
  const int tiles_m = NUM_REL2 / 16;                   // 12
  const int l   = wave / (tiles_m * tiles_n);
  const int rem = wave % (tiles_m * tiles_n);
  const int m0  = (rem / tiles_n) * 16;
  const int n0  = (rem % tiles_n) * 16;

  const int half   = lane >> 4;    // 0: lanes 0-15, 1: lanes 16-31
  const int lane16 = lane & 15;

  // A: row m = m0+lane16 of rel_emb; this lane carries K = k0+2*half, +1
  const float* Arow  = rel_emb + (size_t)(m0 + lane16) * INPUT_DIM;
  // B: col n = n0+lane16 of weight layer l; this lane carries the same K pair
  const float* Bcol  = weight + (size_t)(l * INPUT_DIM) * OUTPUT_DIM + n0 + lane16;

  v8f c = {};
  #pragma unroll 8
  for (int k0 = 0; k0 < INPUT_DIM; k0 += 4) {
    const int ka = k0 + 2 * half;
    v2f a, b;
    a.x = Arow[ka];
    a.y = Arow[ka + 1];
    b.x = Bcol[(size_t)ka * OUTPUT_DIM];
    b.y = Bcol[(size_t)(ka + 1) * OUTPUT_DIM];
    // D = A(16x4) * B(4x16) + C  -> v_wmma_f32_16x16x4_f32
    c = __builtin_amdgcn_wmma_f32_16x16x4_f32(
        /*neg_a=*/false, a, /*neg_b=*/false, b,
        /*c_mod=*/(short)0, c, /*reuse_a=*/false, /*reuse_b=*/false);
  }

  // D layout: VGPR v -> M = v (lanes 0-15) or v+8 (lanes 16-31), N = lane16
  float* Pout = P + ((size_t)(l * NUM_REL2 + m0) * OUTPUT_DIM) + n0 + lane16;
  #pragma unroll
  for (int v = 0; v < 8; ++v) {
    Pout[(size_t)(v + half * 8) * OUTPUT_DIM] = c[v];
  }
}

// ---------------------------------------------------------------------------
// Kernel 2: per-walk scatter. One 256-thread block per walk; thread o adds
// P[0][r0][o]+P[1][r1][o]+P[2][r2][o] atomically into out[seg*256+o].
// The 102 MB accumulator stays resident in the 192 MB L2.
// ---------------------------------------------------------------------------
__global__ void scatter_walks(const int* __restrict__ seg_ids,
                              const int* __restrict__ walk_ids,
                              const float* __restrict__ P,
                              float* __restrict__ out,
                              float* __restrict__ counts) {
  const int w = blockIdx.x;
  const int o = threadIdx.x;              // 0..255
  const int seg = seg_ids[w];
  const int r0  = walk_ids[w * WALK_LEN + 0];
  const int r1  = walk_ids[w * WALK_LEN + 1];
  const int r2  = walk_ids[w * WALK_LEN + 2];

  const float q = P[(size_t)(0 * NUM_REL2 + r0) * OUTPUT_DIM + o]
                + P[(size_t)(1 * NUM_REL2 + r1) * OUTPUT_DIM + o]
                + P[(size_t)(2 * NUM_REL2 + r2) * OUTPUT_DIM + o];

  atomicAdd(out + (size_t)seg * OUTPUT_DIM + o, q);
  if (o == 0) atomicAdd(counts + seg, 1.0f);
}

// ---------------------------------------------------------------------------
// Kernel 3: out[e,o] = relu(out[e,o] / (counts[e]+1) + bias[o])   (in place)
// ---------------------------------------------------------------------------
__global__ void finalize(float* __restrict__ out,
                         const float* __restrict__ counts,
                         const float* __restrict__ bias) {
  const long long idx = (long long)blockIdx.x * blockDim.x + threadIdx.x;
  if (idx >= (long long)NUM_ENTITIES * OUTPUT_DIM) return;
  const int e = (int)(idx >> 8);
  const int o = (int)(idx & (OUTPUT_DIM - 1));
  float v = out[idx] / (counts[e] + 1.0f) + bias[o];
  out[idx] = v > 0.0f ? v : 0.0f;
}

// ---------------------------------------------------------------------------
// Host-side launcher
// Inputs (setup_inputs order):
//   d_in[0]=num_entities (scalar int), d_in[1]=segment_ids (1M int),
//   d_in[2]=walk_ids (3M int), d_in[3]=rel_emb (192*128 f32),
//   d_in[4]=weight (384*256 f32), d_in[5]=bias (256 f32)
// Workspace: P (3*192*256 f32 = 576 KB) then counts (100000 f32).
// ---------------------------------------------------------------------------
extern "C" void kernel_launch(void* const* d_in, const int* in_sizes, int n_in,
                              void* d_out, int out_size, void* d_ws, size_t ws_size,
                              hipStream_t stream) {
  const int*   seg     = (const int*)d_in[1];
  const int*   wids    = (const int*)d_in[2];
  const float* rel_emb = (const float*)d_in[3];
  const float* weight  = (const float*)d_in[4];
  const float* bias    = (const float*)d_in[5];
  float* out = (float*)d_out;

  const size_t P_elems = (size_t)WALK_LEN * NUM_REL2 * OUTPUT_DIM;   // 147456
  float* P      = (float*)d_ws;
  float* counts = (float*)((char*)d_ws + P_elems * sizeof(float));

  // 1) zero accumulators (output + counts) — required every call
  {
    const long long n_out = (long long)NUM_ENTITIES * OUTPUT_DIM;
    zero_f32<<<(unsigned)((n_out + 255) / 256), 256, 0, stream>>>(out, n_out);
    zero_f32<<<(NUM_ENTITIES + 255) / 256, 256, 0, stream>>>(counts, NUM_ENTITIES);
  }

  // 2) projected relation table via fp32 WMMA: 576 waves = 72 blocks x 8 waves
  {
    const int total_waves = WALK_LEN * (NUM_REL2 / 16) * (OUTPUT_DIM / 16); // 576
    const int blocks = (total_waves * 32 + 255) / 256;                      // 72
    project_wmma<<<blocks, 256, 0, stream>>>(rel_emb, weight, P);
  }

  // 3) walk scatter: one block per walk
  scatter_walks<<<NUM_WALKS, 256, 0, stream>>>(seg, wids, P, out, counts);

  // 4) finalize: divide, bias, relu
  {
    const long long n = (long long)NUM_ENTITIES * OUTPUT_DIM;
    finalize<<<(unsigned)((n + 255) / 256), 256, 0, stream>>>(out, counts, bias);
  }
}